// QFCModel_65481071409147
// MI455X (gfx1250) — compile-verified
//
#include <hip/hip_runtime.h>
#include <math.h>

typedef float v2f __attribute__((ext_vector_type(2)));
typedef float v8f __attribute__((ext_vector_type(8)));

#define NQ      8
#define NL      3
#define DIM     256     // 2^NQ
#define SUB     16      // nonzero encoded subspace
#define BATCH   65536
#define NBLK    256     // main-kernel grid
#define BN_EPS  1e-5f

// workspace layout (float offsets)
#define WS_OUTPRE 0                         // [4][BATCH]  channel-major
#define WS_PSUM   (4 * BATCH)               // [4][NBLK]
#define WS_PSQ    (WS_PSUM + 4 * NBLK)      // [4][NBLK]
#define WS_A      (WS_PSQ + 4 * NBLK)       // [4][16][16]  A_k row-major
#define WS_SS     (WS_A + 4 * 256)          // [4][2]  scale, shift

// butterfly add via ds_swizzle immediate (xor mode): imm = (xor<<10) | and 0x1f
template <int IMM>
__device__ __forceinline__ float swz_add(float v) {
  return v + __int_as_float(__builtin_amdgcn_ds_swizzle(__float_as_int(v), IMM));
}
#define SWZ_X1  0x041f
#define SWZ_X2  0x081f
#define SWZ_X4  0x101f
#define SWZ_X8  0x201f
#define SWZ_X16 0x401f

// ---------------------------------------------------------------------------
// Kernel 1: build A_k = U_sub^T diag(g_k) U_sub on device. One block, 256 thr.
// U covers ONLY the 3 parameterized layers (encoding is per-sample).
// ---------------------------------------------------------------------------
__global__ __launch_bounds__(256) void setup_kernel(
    const float* __restrict__ params,   // [3][8]
    const float* __restrict__ W,        // [4][8]
    float* __restrict__ ws) {
  __shared__ float U[DIM * SUB];   // U[s][j]
  __shared__ float st[DIM];
  __shared__ float g[4 * DIM];
  const int t = threadIdx.x;       // 0..255 == state index s

  // g_k[s] = sum_w W[k][w] * (+1 if bit(7-w) of s is 0 else -1)
  for (int k = 0; k < 4; ++k) {
    float acc = 0.f;
    for (int w = 0; w < NQ; ++w) {
      float sgn = ((t >> (7 - w)) & 1) ? -1.f : 1.f;
      acc += W[k * 8 + w] * sgn;
    }
    g[k * DIM + t] = acc;
  }

  // simulate the fixed circuit on the 16 basis states e_{j<<4}
  for (int b = 0; b < SUB; ++b) {
    st[t] = (t == (b << 4)) ? 1.f : 0.f;
    __syncthreads();
    for (int layer = 0; layer < NL; ++layer) {
      for (int q = 0; q < NQ; ++q) {          // RY(params[layer][q]) on wire q
        const int mask = 1 << (7 - q);
        float sn, cs;
        sincosf(params[layer * 8 + q] * 0.5f, &sn, &cs);
        float a0 = st[t & ~mask];
        float a1 = st[t |  mask];
        __syncthreads();
        st[t] = (t & mask) ? (sn * a0 + cs * a1) : (cs * a0 - sn * a1);
        __syncthreads();
      }
      for (int q = 0; q < NQ - 1; ++q) {      // CNOT q -> q+1
        const int mc = 1 << (7 - q), mt = 1 << (6 - q);
        float v = st[(t & mc) ? (t ^ mt) : t];
        __syncthreads();
        st[t] = v;
        __syncthreads();
      }
    }
    U[t * SUB + b] = st[t];
    __syncthreads();
  }

  // A_k[j1][j2] = sum_s g_k[s] U[s][j1] U[s][j2]
  const int j1 = t >> 4, j2 = t & 15;
  for (int k = 0; k < 4; ++k) {
    float acc = 0.f;
    for (int s = 0; s < DIM; ++s)
      acc += g[k * DIM + s] * U[s * SUB + j1] * U[s * SUB + j2];
    ws[WS_A + k * 256 + j1 * 16 + j2] = acc;
  }
}

// ---------------------------------------------------------------------------
// Kernel 2: pooling + v + quadratic forms. D = A_k x V^T via WMMA so the
// final dot reduction is 8 in-register FMAs + one xor-16 combine per output.
// 256 blocks x 256 threads, one sample per thread, wave = two 16-sample tiles.
// ---------------------------------------------------------------------------
__global__ __launch_bounds__(256) void main_kernel(
    const float* __restrict__ x,       // [B][144]
    const float* __restrict__ bias4,   // [4]
    float* __restrict__ ws) {
  __shared__ float vt[256 * 17];       // V, stride 17 (bank-conflict free)
  __shared__ float wsum[32], wsq[32];  // [wave][k]
  const int tid  = threadIdx.x;
  const int lane = tid & 31;
  const int wave = tid >> 5;
  const int sample = blockIdx.x * 256 + tid;

  // ---- 2x2 average pool of 12x12 (rows are 48B => float4 aligned) ----
  const float4* xr = (const float4*)(x + (size_t)sample * 144);
  float p0 = 0.f, p1 = 0.f, p2 = 0.f, p3 = 0.f;
#pragma unroll
  for (int r = 0; r < 12; ++r) {
    float4 a = xr[r * 3 + 0];
    float4 b = xr[r * 3 + 1];
    float4 c = xr[r * 3 + 2];
    float left  = a.x + a.y + a.z + a.w + b.x + b.y;
    float right = b.z + b.w + c.x + c.y + c.z + c.w;
    if (r < 6) { p0 += left; p1 += right; } else { p2 += left; p3 += right; }
  }
  const float hs = (1.f / 36.f) * 0.5f;   // mean then half-angle
  float s0, c0, s1, c1, s2, c2, s3, c3;
  sincosf(p0 * hs, &s0, &c0);
  sincosf(p1 * hs, &s1, &c1);
  sincosf(p2 * hs, &s2, &c2);
  sincosf(p3 * hs, &s3, &c3);

  // v[j] = f0(b0)*f1(b1)*f2(b2)*f3(b3), j = b0b1b2b3 (wire0 = MSB)
  float f01[4] = {c0 * c1, c0 * s1, s0 * c1, s0 * s1};
  float f23[4] = {c2 * c3, c2 * s3, s2 * c3, s2 * s3};
#pragma unroll
  for (int j = 0; j < 16; ++j)
    vt[tid * 17 + j] = f01[j >> 2] * f23[j & 3];
  __syncthreads();

  // ---- A operands: A_k in WMMA A layout (M=j1 rows, K=j2), resident ----
  const int koff = (lane < 16) ? 0 : 2;     // K half select
  const int mrow = lane & 15;               // M index (j1) / N index (sample)
  const float* Amat = ws + WS_A;
  v2f aop[4][4];
#pragma unroll
  for (int k = 0; k < 4; ++k)
#pragma unroll
    for (int kk = 0; kk < 4; ++kk) {
      aop[k][kk].x = Amat[k * 256 + mrow * 16 + 4 * kk + koff];
      aop[k][kk].y = Amat[k * 256 + mrow * 16 + 4 * kk + koff + 1];
    }

  float bk[4];
#pragma unroll
  for (int k = 0; k < 4; ++k) bk[k] = bias4[k];

  float psum[4] = {0.f, 0.f, 0.f, 0.f};
  float psq[4]  = {0.f, 0.f, 0.f, 0.f};
  const v8f vzero = {0.f, 0.f, 0.f, 0.f, 0.f, 0.f, 0.f, 0.f};

#pragma unroll
  for (int T = 0; T < 2; ++T) {
    const int rowbase = wave * 32 + T * 16;
    const int myrow   = rowbase + mrow;     // this lane's sample row in vt

    v8f acc[4] = {vzero, vzero, vzero, vzero};
#pragma unroll
    for (int kk = 0; kk < 4; ++kk) {
      v2f bop;                              // B layout: V^T chunk (K=j2, N=m)
      const int kb = 4 * kk + koff;
      bop.x = vt[myrow * 17 + kb];
      bop.y = vt[myrow * 17 + kb + 1];
#pragma unroll
      for (int k = 0; k < 4; ++k)
        acc[k] = __builtin_amdgcn_wmma_f32_16x16x4_f32(
            false, aop[k][kk], false, bop, (short)0, acc[k], false, false);
    }

    // acc[k][r] = (A_k v_m)[j1], j1 = r + (lane<16 ? 0 : 8), m = lane&15
    const int cb = (lane < 16) ? 0 : 8;
    float vr[8];
#pragma unroll
    for (int r = 0; r < 8; ++r)
      vr[r] = vt[myrow * 17 + cb + r];

#pragma unroll
    for (int k = 0; k < 4; ++k) {
      float d = 0.f;
#pragma unroll
      for (int r = 0; r < 8; ++r)
        d = fmaf(acc[k][r], vr[r], d);      // half of the j1-dot, in-register
      d = swz_add<SWZ_X16>(d);              // combine lane halves
      const float val = d + bk[k];
      if (lane < 16) {                      // 16-wide contiguous store
        ws[WS_OUTPRE + k * BATCH + blockIdx.x * 256 + rowbase + mrow] = val;
        psum[k] += val;
        psq[k]  += val * val;
      }
    }
  }

  // ---- deterministic block partial sums (fixed order, no FP atomics) ----
#pragma unroll
  for (int k = 0; k < 4; ++k) {
    float a = psum[k], b = psq[k];
    a = swz_add<SWZ_X1>(a);   b = swz_add<SWZ_X1>(b);
    a = swz_add<SWZ_X2>(a);   b = swz_add<SWZ_X2>(b);
    a = swz_add<SWZ_X4>(a);   b = swz_add<SWZ_X4>(b);
    a = swz_add<SWZ_X8>(a);   b = swz_add<SWZ_X8>(b);
    a = swz_add<SWZ_X16>(a);  b = swz_add<SWZ_X16>(b);
    if (lane == 0) { wsum[wave * 4 + k] = a; wsq[wave * 4 + k] = b; }
  }
  __syncthreads();
  if (tid < 4) {
    float a = 0.f, b = 0.f;
    for (int wv = 0; wv < 8; ++wv) { a += wsum[wv * 4 + tid]; b += wsq[wv * 4 + tid]; }
    ws[WS_PSUM + tid * NBLK + blockIdx.x] = a;
    ws[WS_PSQ  + tid * NBLK + blockIdx.x] = b;
  }
}

// ---------------------------------------------------------------------------
// Kernel 3: reduce partials -> per-channel scale/shift (1 tiny block)
// ---------------------------------------------------------------------------
__global__ void finalize_kernel(const float* __restrict__ gamma,
                                const float* __restrict__ beta,
                                float* __restrict__ ws) {
  const int k = threadIdx.x;
  if (k < 4) {
    float s = 0.f, q = 0.f;
    for (int i = 0; i < NBLK; ++i) {
      s += ws[WS_PSUM + k * NBLK + i];
      q += ws[WS_PSQ  + k * NBLK + i];
    }
    const float mean  = s * (1.f / BATCH);
    const float var   = q * (1.f / BATCH) - mean * mean;
    const float scale = gamma[k] * rsqrtf(var + BN_EPS);
    ws[WS_SS + 2 * k]     = scale;
    ws[WS_SS + 2 * k + 1] = beta[k] - mean * scale;
  }
}

// ---------------------------------------------------------------------------
// Kernel 4: apply batch-norm, write (B,4) row-major output
// ---------------------------------------------------------------------------
__global__ __launch_bounds__(256) void normalize_kernel(
    const float* __restrict__ ws, float* __restrict__ out) {
  const int s = blockIdx.x * 256 + threadIdx.x;
  float ss[8];
#pragma unroll
  for (int i = 0; i < 8; ++i) ss[i] = ws[WS_SS + i];
  float4 o;
  o.x = ws[WS_OUTPRE + 0 * BATCH + s] * ss[0] + ss[1];
  o.y = ws[WS_OUTPRE + 1 * BATCH + s] * ss[2] + ss[3];
  o.z = ws[WS_OUTPRE + 2 * BATCH + s] * ss[4] + ss[5];
  o.w = ws[WS_OUTPRE + 3 * BATCH + s] * ss[6] + ss[7];
  ((float4*)out)[s] = o;
}

extern "C" void kernel_launch(void* const* d_in, const int* in_sizes, int n_in,
                              void* d_out, int out_size, void* d_ws, size_t ws_size,
                              hipStream_t stream) {
  const float* x      = (const float*)d_in[0];
  const float* params = (const float*)d_in[1];
  const float* W      = (const float*)d_in[2];
  const float* b      = (const float*)d_in[3];
  const float* gamma  = (const float*)d_in[4];
  const float* beta   = (const float*)d_in[5];
  float* ws  = (float*)d_ws;
  float* out = (float*)d_out;

  setup_kernel<<<1, 256, 0, stream>>>(params, W, ws);
  main_kernel<<<NBLK, 256, 0, stream>>>(x, b, ws);
  finalize_kernel<<<1, 64, 0, stream>>>(gamma, beta, ws);
  normalize_kernel<<<BATCH / 256, 256, 0, stream>>>(ws, out);
}